// ConfEstModule_33397665694039
// MI455X (gfx1250) — compile-verified
//
#include <hip/hip_runtime.h>
#include <math.h>

// ---------------------------------------------------------------------------
// ConfEstModule on MI455X (gfx1250):
//   6D radius-KNN(64) -> masked mean of x  -> x_centers   (per 3 scales)
//   3D radius-KNN(64) -> masked mean of ||c_i - c_j||     (per 3 scales)
//   sigmoid(w . msg + b)
//
// Gram matrices via V_WMMA_F32_16X16X4_F32 (f32 WMMA: selection is rank-
// sensitive, f16 would perturb the top-64 ordering). 16x2048 d^2 tile lives
// in LDS; top-64 threshold by uint-bisection with b128 LDS scans; per-scale
// masked reductions re-scan the LDS tile. x staged to LDS with
// GLOBAL_LOAD_ASYNC_TO_LDS_B32 (ASYNCcnt) when available.
// ---------------------------------------------------------------------------

typedef __attribute__((ext_vector_type(2))) float v2f;
typedef __attribute__((ext_vector_type(8))) float v8f;

#define MPTS    2048
#define NBATCH  8
#define KNN     64
#define TILE    16
#define NWAVES  16
#define THREADS 512

struct __align__(16) F4 { float a, b, c, d; };

#if __has_builtin(__builtin_amdgcn_global_load_async_to_lds_b32)
#define HAVE_ASYNC_LDS 1
#else
#define HAVE_ASYNC_LDS 0
#endif

__device__ __forceinline__ void async_cp_f32(float* lds_dst, const float* gsrc) {
#if HAVE_ASYNC_LDS
    __builtin_amdgcn_global_load_async_to_lds_b32(
        (__attribute__((address_space(1))) int*)gsrc,
        (__attribute__((address_space(3))) int*)lds_dst, 0, 0);
#else
    *lds_dst = *gsrc;
#endif
}

__device__ __forceinline__ void async_wait_all() {
#if HAVE_ASYNC_LDS
#if __has_builtin(__builtin_amdgcn_s_wait_asynccnt)
    __builtin_amdgcn_s_wait_asynccnt(0);
#else
    asm volatile("s_wait_asynccnt 0x0" ::: "memory");
#endif
#endif
}

// Wave32 butterfly sum (all lanes end with the total).
__device__ __forceinline__ float waveSumF(float v) {
#pragma unroll
    for (int off = 16; off; off >>= 1) v += __shfl_xor(v, off);
    return v;
}
__device__ __forceinline__ int waveSumI(int v) {
#pragma unroll
    for (int off = 16; off; off >>= 1) v += __shfl_xor(v, off);
    return v;
}

// KNN-th smallest of rowp[0..MPTS) (all >= 0, all < 16.0) by bisection on the
// float bit pattern; one wave cooperates, result on all lanes. b128 LDS scans.
__device__ __forceinline__ float kth_smallest64(const float* __restrict__ rowp, int lane) {
    const F4* row4 = (const F4*)rowp;
    unsigned lo = 0u, hi = 0x41800000u; // 16.0f bits (d^2 < 16 guaranteed)
    while (lo < hi) {
        unsigned mid = (lo + hi) >> 1;
        int cnt = 0;
        for (int t = lane; t < MPTS / 4; t += 32) {
            F4 v = row4[t];
            cnt += (__float_as_uint(v.a) <= mid) ? 1 : 0;
            cnt += (__float_as_uint(v.b) <= mid) ? 1 : 0;
            cnt += (__float_as_uint(v.c) <= mid) ? 1 : 0;
            cnt += (__float_as_uint(v.d) <= mid) ? 1 : 0;
        }
        cnt = waveSumI(cnt);
        if (cnt >= KNN) hi = mid; else lo = mid + 1;
    }
    return __uint_as_float(hi);
}

// ---------------------------------------------------------------------------
// Kernel 1: 6D graph -> x_centers[scale][b][i][3]
// ---------------------------------------------------------------------------
extern "C" __global__ __launch_bounds__(THREADS)
void conf_k1_centers(const float* __restrict__ x, const float* __restrict__ pos,
                     float* __restrict__ centers) {
    __shared__ float d2s[TILE * MPTS];   // 128 KB distance tile
    __shared__ float xs[MPTS * 3];       // 24 KB x stage (phase-D re-scans)
    __shared__ float sqi[TILE];          // row 6D norms

    const int b    = blockIdx.y;
    const int i0   = blockIdx.x * TILE;
    const int tid  = threadIdx.x;
    const int lane = tid & 31;
    const int wv   = tid >> 5;
    const int l16  = lane & 15;
    const int hi16 = lane >> 4;
    const size_t b3 = (size_t)b * MPTS * 3;

    const float R6sq[3] = {0.15f * 0.15f, 0.25f * 0.25f, 0.45f * 0.45f};

    // Async-stage x into LDS; compute the 16 row norms.
    for (int j = tid; j < MPTS; j += THREADS) {
        async_cp_f32(&xs[3 * j + 0], &x[b3 + 3 * j + 0]);
        async_cp_f32(&xs[3 * j + 1], &x[b3 + 3 * j + 1]);
        async_cp_f32(&xs[3 * j + 2], &x[b3 + 3 * j + 2]);
    }
    if (tid < TILE) {
        const int i = i0 + tid;
        float p0 = pos[b3 + 3 * i + 0], p1 = pos[b3 + 3 * i + 1], p2 = pos[b3 + 3 * i + 2];
        float q0 = x[b3 + 3 * i + 0],   q1 = x[b3 + 3 * i + 1],   q2 = x[b3 + 3 * i + 2];
        sqi[tid] = p0 * p0 + p1 * p1 + p2 * p2 + q0 * q0 + q1 * q1 + q2 * q2;
    }
    async_wait_all();
    __syncthreads();

    // A fragment (rows i0..i0+15, K padded 6->8). 16x4 f32 A layout:
    // lanes 0-15 hold K={0,1}, lanes 16-31 hold K={2,3}. Branchless selects.
    v2f aA, aB;
    {
        const int ar = i0 + l16;
        float p0 = pos[b3 + 3 * ar + 0], p1 = pos[b3 + 3 * ar + 1], p2 = pos[b3 + 3 * ar + 2];
        float q0 = x[b3 + 3 * ar + 0],   q1 = x[b3 + 3 * ar + 1],   q2 = x[b3 + 3 * ar + 2];
        aA.x = hi16 ? p2 : p0;
        aA.y = hi16 ? q0 : p1;
        aB.x = hi16 ? 0.f : q1;
        aB.y = hi16 ? 0.f : q2;
    }

    // Each wave computes 8 of the 128 16-wide column tiles.
    for (int ct = wv; ct < MPTS / 16; ct += NWAVES) {
        const int jb = ct * 16;
        const int jn = jb + l16;
        float p0 = pos[b3 + 3 * jn + 0], p1 = pos[b3 + 3 * jn + 1], p2 = pos[b3 + 3 * jn + 2];
        float q0 = x[b3 + 3 * jn + 0],   q1 = x[b3 + 3 * jn + 1],   q2 = x[b3 + 3 * jn + 2];
        const float sqcol = p0 * p0 + p1 * p1 + p2 * p2 + q0 * q0 + q1 * q1 + q2 * q2;
        v2f bA, bB;
        bA.x = hi16 ? p2 : p0;
        bA.y = hi16 ? q0 : p1;
        bB.x = hi16 ? 0.f : q1;
        bB.y = hi16 ? 0.f : q2;

        v8f c = {};
        c = __builtin_amdgcn_wmma_f32_16x16x4_f32(false, aA, false, bA, (short)0, c, false, false);
        c = __builtin_amdgcn_wmma_f32_16x16x4_f32(false, aB, false, bB, (short)0, c, false, false);

#pragma unroll
        for (int r = 0; r < 8; ++r) {
            const int row = r + (hi16 ? 8 : 0);
            float d = sqi[row] + sqcol - 2.0f * c[r];
            d2s[row * MPTS + jb + l16] = fmaxf(d, 0.0f);
        }
    }
    __syncthreads();

    // One wave per row: top-64 threshold, then 3-scale masked mean of x (LDS).
    const float* rowp = &d2s[wv * MPTS];
    const float t64 = kth_smallest64(rowp, lane);

    const float thr0 = fminf(t64, R6sq[0]);
    const float thr1 = fminf(t64, R6sq[1]);
    const float thr2 = fminf(t64, R6sq[2]);   // thr0 <= thr1 <= thr2

    float s00 = 0.f, s01 = 0.f, s02 = 0.f;
    float s10 = 0.f, s11 = 0.f, s12 = 0.f;
    float s20 = 0.f, s21 = 0.f, s22 = 0.f;
    int c0 = 0, c1 = 0, c2 = 0;
    const F4* row4 = (const F4*)rowp;
    for (int t = lane; t < MPTS / 4; t += 32) {
        F4 v = row4[t];
        float dv[4] = {v.a, v.b, v.c, v.d};
#pragma unroll
        for (int e = 0; e < 4; ++e) {
            float d = dv[e];
            if (d <= thr2) {
                const int j = 4 * t + e;
                float q0 = xs[3 * j + 0], q1 = xs[3 * j + 1], q2 = xs[3 * j + 2];
                s20 += q0; s21 += q1; s22 += q2; c2++;
                if (d <= thr1) {
                    s10 += q0; s11 += q1; s12 += q2; c1++;
                    if (d <= thr0) { s00 += q0; s01 += q1; s02 += q2; c0++; }
                }
            }
        }
    }
    s00 = waveSumF(s00); s01 = waveSumF(s01); s02 = waveSumF(s02);
    s10 = waveSumF(s10); s11 = waveSumF(s11); s12 = waveSumF(s12);
    s20 = waveSumF(s20); s21 = waveSumF(s21); s22 = waveSumF(s22);
    c0 = waveSumI(c0); c1 = waveSumI(c1); c2 = waveSumI(c2);

    if (lane == 0) {
        const int i = i0 + wv;
        float i0c = 1.0f / (float)c0, i1c = 1.0f / (float)c1, i2c = 1.0f / (float)c2;
        size_t o0 = ((size_t)(0 * NBATCH + b) * MPTS + i) * 3;
        size_t o1 = ((size_t)(1 * NBATCH + b) * MPTS + i) * 3;
        size_t o2 = ((size_t)(2 * NBATCH + b) * MPTS + i) * 3;
        centers[o0 + 0] = s00 * i0c; centers[o0 + 1] = s01 * i0c; centers[o0 + 2] = s02 * i0c;
        centers[o1 + 0] = s10 * i1c; centers[o1 + 1] = s11 * i1c; centers[o1 + 2] = s12 * i1c;
        centers[o2 + 0] = s20 * i2c; centers[o2 + 1] = s21 * i2c; centers[o2 + 2] = s22 * i2c;
    }
}

// ---------------------------------------------------------------------------
// Kernel 2: 3D graph -> msg means -> sigmoid head -> out[b*M + i]
// ---------------------------------------------------------------------------
extern "C" __global__ __launch_bounds__(THREADS)
void conf_k2_out(const float* __restrict__ pos, const float* __restrict__ centers,
                 const float* __restrict__ w, const float* __restrict__ bias,
                 float* __restrict__ out) {
    __shared__ float d2s[TILE * MPTS];   // 128 KB distance tile
    __shared__ float sqi[TILE];

    const int b    = blockIdx.y;
    const int i0   = blockIdx.x * TILE;
    const int tid  = threadIdx.x;
    const int lane = tid & 31;
    const int wv   = tid >> 5;
    const int l16  = lane & 15;
    const int hi16 = lane >> 4;
    const size_t b3 = (size_t)b * MPTS * 3;

    const float R3sq[3] = {0.1f * 0.1f, 0.2f * 0.2f, 0.4f * 0.4f};

    if (tid < TILE) {
        const int i = i0 + tid;
        float p0 = pos[b3 + 3 * i + 0], p1 = pos[b3 + 3 * i + 1], p2 = pos[b3 + 3 * i + 2];
        sqi[tid] = p0 * p0 + p1 * p1 + p2 * p2;
    }
    __syncthreads();

    // A fragment: 3D, K padded 3->4 (single WMMA k-step), branchless.
    v2f aA;
    {
        const int ar = i0 + l16;
        float p0 = pos[b3 + 3 * ar + 0], p1 = pos[b3 + 3 * ar + 1], p2 = pos[b3 + 3 * ar + 2];
        aA.x = hi16 ? p2 : p0;
        aA.y = hi16 ? 0.f : p1;
    }

    for (int ct = wv; ct < MPTS / 16; ct += NWAVES) {
        const int jb = ct * 16;
        const int jn = jb + l16;
        float p0 = pos[b3 + 3 * jn + 0], p1 = pos[b3 + 3 * jn + 1], p2 = pos[b3 + 3 * jn + 2];
        const float sqcol = p0 * p0 + p1 * p1 + p2 * p2;
        v2f bA;
        bA.x = hi16 ? p2 : p0;
        bA.y = hi16 ? 0.f : p1;

        v8f c = {};
        c = __builtin_amdgcn_wmma_f32_16x16x4_f32(false, aA, false, bA, (short)0, c, false, false);

#pragma unroll
        for (int r = 0; r < 8; ++r) {
            const int row = r + (hi16 ? 8 : 0);
            float d = sqi[row] + sqcol - 2.0f * c[r];
            d2s[row * MPTS + jb + l16] = fmaxf(d, 0.0f);
        }
    }
    __syncthreads();

    const float* rowp = &d2s[wv * MPTS];
    const float t64 = kth_smallest64(rowp, lane);

    const float thr0 = fminf(t64, R3sq[0]);
    const float thr1 = fminf(t64, R3sq[1]);
    const float thr2 = fminf(t64, R3sq[2]);

    const int i = i0 + wv;
    const size_t cb0 = (size_t)(0 * NBATCH + b) * MPTS * 3;
    const size_t cb1 = (size_t)(1 * NBATCH + b) * MPTS * 3;
    const size_t cb2 = (size_t)(2 * NBATCH + b) * MPTS * 3;
    const float ci00 = centers[cb0 + 3 * i + 0], ci01 = centers[cb0 + 3 * i + 1], ci02 = centers[cb0 + 3 * i + 2];
    const float ci10 = centers[cb1 + 3 * i + 0], ci11 = centers[cb1 + 3 * i + 1], ci12 = centers[cb1 + 3 * i + 2];
    const float ci20 = centers[cb2 + 3 * i + 0], ci21 = centers[cb2 + 3 * i + 1], ci22 = centers[cb2 + 3 * i + 2];

    float sum0 = 0.f, sum1 = 0.f, sum2 = 0.f;
    int c0 = 0, c1 = 0, c2 = 0;
    const F4* row4 = (const F4*)rowp;
    for (int t = lane; t < MPTS / 4; t += 32) {
        F4 v = row4[t];
        float dv[4] = {v.a, v.b, v.c, v.d};
        __builtin_prefetch(&centers[cb2 + 3 * (4 * t)], 0, 1);  // global_prefetch_b8
#pragma unroll
        for (int e = 0; e < 4; ++e) {
            float d = dv[e];
            if (d <= thr2) {
                const int j = 4 * t + e;
                {
                    float dx = ci20 - centers[cb2 + 3 * j + 0];
                    float dy = ci21 - centers[cb2 + 3 * j + 1];
                    float dz = ci22 - centers[cb2 + 3 * j + 2];
                    sum2 += sqrtf(fmaxf(dx * dx + dy * dy + dz * dz, 1e-12f)); c2++;
                }
                if (d <= thr1) {
                    float dx = ci10 - centers[cb1 + 3 * j + 0];
                    float dy = ci11 - centers[cb1 + 3 * j + 1];
                    float dz = ci12 - centers[cb1 + 3 * j + 2];
                    sum1 += sqrtf(fmaxf(dx * dx + dy * dy + dz * dz, 1e-12f)); c1++;
                    if (d <= thr0) {
                        float ex = ci00 - centers[cb0 + 3 * j + 0];
                        float ey = ci01 - centers[cb0 + 3 * j + 1];
                        float ez = ci02 - centers[cb0 + 3 * j + 2];
                        sum0 += sqrtf(fmaxf(ex * ex + ey * ey + ez * ez, 1e-12f)); c0++;
                    }
                }
            }
        }
    }
    sum0 = waveSumF(sum0); sum1 = waveSumF(sum1); sum2 = waveSumF(sum2);
    c0 = waveSumI(c0); c1 = waveSumI(c1); c2 = waveSumI(c2);

    if (lane == 0) {
        float m0 = sum0 / (float)c0;
        float m1 = sum1 / (float)c1;
        float m2 = sum2 / (float)c2;
        float z = w[0] * m0 + w[1] * m1 + w[2] * m2 + bias[0];
        out[(size_t)b * MPTS + i] = 1.0f / (1.0f + expf(-z));
    }
}

// ---------------------------------------------------------------------------
extern "C" void kernel_launch(void* const* d_in, const int* in_sizes, int n_in,
                              void* d_out, int out_size, void* d_ws, size_t ws_size,
                              hipStream_t stream) {
    const float* x   = (const float*)d_in[0];   // [8,2048,3]
    const float* pos = (const float*)d_in[1];   // [8,2048,3]
    const float* w   = (const float*)d_in[2];   // [1,3]
    const float* b   = (const float*)d_in[3];   // [1]
    float* out       = (float*)d_out;           // [16384]
    float* centers   = (float*)d_ws;            // [3,8,2048,3] floats = 576 KB

    dim3 grid(MPTS / TILE, NBATCH);             // 128 x 8 tiles
    conf_k1_centers<<<grid, THREADS, 0, stream>>>(x, pos, centers);
    conf_k2_out<<<grid, THREADS, 0, stream>>>(pos, centers, w, b, out);
}